// NeuralNetwork_6519760355912
// MI455X (gfx1250) — compile-verified
//
#include <hip/hip_runtime.h>

typedef __attribute__((ext_vector_type(2))) float v2f;
typedef __attribute__((ext_vector_type(4))) float v4f;
typedef __attribute__((ext_vector_type(8))) float v8f;

#define D_IN  2048
#define D_H   1024
#define D_OUT 1000
#define BATCH 100
#define M_PAD 112   // 7 tiles of 16

// workspace layout (float offsets)
#define OFF_H   0            // 112*1024      = 114688
#define OFF_M1  114688       // 100*1024      = 102400
#define OFF_M2  217088       // 100*1000      = 100000
#define OFF_P1  317088       // 8*100*1024    = 819200
#define OFF_P2  1136288      // 8*100*1000    = 800000
// total floats = 1,936,288  (~7.8 MB)

// ---- streaming partial reduction of b1 over axis 1 (8-way split, b128 NT loads) ----
// grid: (BATCH, 8), block: 256.  Each thread owns 4 consecutive columns (float4),
// 256 threads * 4 = 1024 = D_H, and sums 256 rows of its split.
__global__ void __launch_bounds__(256) partial_b1(const float* __restrict__ b1,
                                                  float* __restrict__ p1) {
    const int o4 = threadIdx.x;          // 0..255 -> columns 4*o4..4*o4+3
    const int b  = blockIdx.x;           // 0..99
    const int s  = blockIdx.y;           // 0..7
    const v4f* src = (const v4f*)(b1 + (size_t)b * D_IN * D_H
                                     + (size_t)(s * 256) * D_H) + o4;
    v4f acc = {};
    #pragma unroll 8
    for (int i = 0; i < 256; ++i) {
        v4f v = __builtin_nontemporal_load(src + (size_t)i * (D_H / 4));
        acc += v;
    }
    *((v4f*)(p1 + (size_t)s * (BATCH * D_H) + (size_t)b * D_H) + o4) = acc;
}

// ---- streaming partial reduction of b2 over axis 1 (8-way split, b128 NT loads) ----
// grid: (BATCH, 8), block: 256 (250 active).  D_OUT = 1000 = 250 float4s.
__global__ void __launch_bounds__(256) partial_b2(const float* __restrict__ b2,
                                                  float* __restrict__ p2) {
    const int o4 = threadIdx.x;          // 0..255, active if < 250
    const int b  = blockIdx.x;
    const int s  = blockIdx.y;
    if (o4 >= D_OUT / 4) return;
    const v4f* src = (const v4f*)(b2 + (size_t)b * D_H * D_OUT
                                     + (size_t)(s * 128) * D_OUT) + o4;
    v4f acc = {};
    #pragma unroll 8
    for (int i = 0; i < 128; ++i) {
        v4f v = __builtin_nontemporal_load(src + (size_t)i * (D_OUT / 4));
        acc += v;
    }
    *((v4f*)(p2 + (size_t)s * (BATCH * D_OUT) + (size_t)b * D_OUT) + o4) = acc;
}

// ---- collapse the 8 partials into the mean (deterministic, no atomics) ----
// n4 = number of float4 elements (n/4).
__global__ void __launch_bounds__(256) finalize_mean(const float* __restrict__ p,
                                                     float* __restrict__ m,
                                                     int n4, float scale) {
    const int idx = blockIdx.x * 256 + threadIdx.x;
    if (idx >= n4) return;
    v4f acc = {};
    #pragma unroll
    for (int s = 0; s < 8; ++s) acc += *((const v4f*)p + (size_t)s * n4 + idx);
    *((v4f*)m + idx) = acc * scale;
}

// ---- GEMM1: h = relu(x @ W1 / D_IN + mean1), stored padded to 112 rows ----
__global__ void __launch_bounds__(32) gemm1_wmma(const float* __restrict__ x,
                                                 const float* __restrict__ W1,
                                                 const float* __restrict__ mean1,
                                                 float* __restrict__ h) {
    const int lane  = threadIdx.x;
    const int n0    = blockIdx.x * 16;
    const int m0    = blockIdx.y * 16;
    const int lhalf = lane >> 4;       // 0: K=k0,k0+1  1: K=k0+2,k0+3
    const int lmod  = lane & 15;

    const int arow  = m0 + lmod;                       // A row (may exceed 99)
    const int srow  = arow < BATCH ? arow : 0;         // clamp address; junk rows discarded
    const float* aptr = x  + (size_t)srow * D_IN + 2 * lhalf;
    const float* bptr = W1 + (size_t)(2 * lhalf) * D_H + n0 + lmod;

    v8f c = {};
    #pragma unroll 4
    for (int k0 = 0; k0 < D_IN; k0 += 4) {
        v2f a;
        a.x = aptr[k0];
        a.y = aptr[k0 + 1];
        v2f b;
        b.x = bptr[(size_t)k0 * D_H];
        b.y = bptr[(size_t)(k0 + 1) * D_H];
        c = __builtin_amdgcn_wmma_f32_16x16x4_f32(false, a, false, b,
                                                  (short)0, c, false, false);
    }

    const int col = n0 + lmod;
    if (m0 + 16 <= BATCH) {
        // fast path: whole tile in-bounds, no per-element EXEC churn
        #pragma unroll
        for (int r = 0; r < 8; ++r) {
            const int orow = m0 + r + 8 * lhalf;
            float v = c[r] * (1.0f / D_IN) + mean1[(size_t)orow * D_H + col];
            h[(size_t)orow * D_H + col] = v > 0.f ? v : 0.f;
        }
    } else {
        #pragma unroll
        for (int r = 0; r < 8; ++r) {
            const int orow = m0 + r + 8 * lhalf;       // < 112 always
            float v = 0.f;
            if (orow < BATCH) {
                v = c[r] * (1.0f / D_IN) + mean1[(size_t)orow * D_H + col];
                v = v > 0.f ? v : 0.f;
            }
            h[(size_t)orow * D_H + col] = v;           // zero-fill padded rows
        }
    }
}

// ---- GEMM2: out = h @ W2 / D_H + mean2 ----
__global__ void __launch_bounds__(32) gemm2_wmma(const float* __restrict__ h,
                                                 const float* __restrict__ W2,
                                                 const float* __restrict__ mean2,
                                                 float* __restrict__ out) {
    const int lane  = threadIdx.x;
    const int n0    = blockIdx.x * 16;
    const int m0    = blockIdx.y * 16;
    const int lhalf = lane >> 4;
    const int lmod  = lane & 15;

    const int arow = m0 + lmod;                        // < 112, h is padded
    const int col  = n0 + lmod;
    const int ccol = col < D_OUT ? col : 0;            // clamp B address; discarded at store
    const float* aptr = h  + (size_t)arow * D_H + 2 * lhalf;
    const float* bptr = W2 + (size_t)(2 * lhalf) * D_OUT + ccol;

    v8f c = {};
    #pragma unroll 4
    for (int k0 = 0; k0 < D_H; k0 += 4) {
        v2f a;
        a.x = aptr[k0];
        a.y = aptr[k0 + 1];
        v2f b;
        b.x = bptr[(size_t)k0 * D_OUT];
        b.y = bptr[(size_t)(k0 + 1) * D_OUT];
        c = __builtin_amdgcn_wmma_f32_16x16x4_f32(false, a, false, b,
                                                  (short)0, c, false, false);
    }

    if (m0 + 16 <= BATCH && n0 + 16 <= D_OUT) {
        // fast path: interior tile
        #pragma unroll
        for (int r = 0; r < 8; ++r) {
            const int orow = m0 + r + 8 * lhalf;
            out[(size_t)orow * D_OUT + col] =
                c[r] * (1.0f / D_H) + mean2[(size_t)orow * D_OUT + col];
        }
    } else {
        #pragma unroll
        for (int r = 0; r < 8; ++r) {
            const int orow = m0 + r + 8 * lhalf;
            if (orow < BATCH && col < D_OUT) {
                out[(size_t)orow * D_OUT + col] =
                    c[r] * (1.0f / D_H) + mean2[(size_t)orow * D_OUT + col];
            }
        }
    }
}

extern "C" void kernel_launch(void* const* d_in, const int* in_sizes, int n_in,
                              void* d_out, int out_size, void* d_ws, size_t ws_size,
                              hipStream_t stream) {
    const float* x  = (const float*)d_in[0];
    const float* W1 = (const float*)d_in[1];
    const float* b1 = (const float*)d_in[2];
    const float* W2 = (const float*)d_in[3];
    const float* b2 = (const float*)d_in[4];
    float* out = (float*)d_out;
    float* ws  = (float*)d_ws;

    float* h  = ws + OFF_H;
    float* m1 = ws + OFF_M1;
    float* m2 = ws + OFF_M2;
    float* p1 = ws + OFF_P1;
    float* p2 = ws + OFF_P2;

    // bandwidth-bound bias reductions: 1.25 GB of NT b128 streaming (the roofline term)
    partial_b1<<<dim3(BATCH, 8), 256, 0, stream>>>(b1, p1);
    partial_b2<<<dim3(BATCH, 8), 256, 0, stream>>>(b2, p2);
    finalize_mean<<<(BATCH * D_H / 4 + 255) / 256, 256, 0, stream>>>(p1, m1, BATCH * D_H / 4, 1.0f / D_IN);
    finalize_mean<<<(BATCH * D_OUT / 4 + 255) / 256, 256, 0, stream>>>(p2, m2, BATCH * D_OUT / 4, 1.0f / D_H);

    // WMMA f32 GEMMs with fused scale + bias-mean (+ relu)
    gemm1_wmma<<<dim3(D_H / 16, M_PAD / 16), 32, 0, stream>>>(x, W1, m1, h);
    gemm2_wmma<<<dim3((D_OUT + 15) / 16, M_PAD / 16), 32, 0, stream>>>(h, W2, m2, out);
}